// GAT_85203561218533
// MI455X (gfx1250) — compile-verified
//
#include <hip/hip_runtime.h>

typedef float v2f __attribute__((ext_vector_type(2)));
typedef float v8f __attribute__((ext_vector_type(8)));

#define IN_C 256
#define HID 64
#define HEADS 4
#define OUT_C 64
#define NEG_SLOPE 0.2f

// ---------- helpers ----------
__device__ __forceinline__ unsigned flipf(float f) {
    unsigned u = __float_as_uint(f);
    return (u & 0x80000000u) ? ~u : (u | 0x80000000u);
}
__device__ __forceinline__ float unflipf(unsigned u) {
    u = (u & 0x80000000u) ? (u & 0x7FFFFFFFu) : ~u;
    return __uint_as_float(u);
}
__device__ __forceinline__ void resolve_edge(const long long* ei, long long E,
                                             long long e, int& s, int& d) {
    if (e < E) { s = (int)ei[e]; d = (int)ei[E + e]; }
    else       { s = d = (int)(e - E); }
}

// ---------- FP32 WMMA GEMM: C[M,NC] = A[M,K] @ B[K,NC] ----------
// One wave computes a 16x64 strip (4 accumulators): A fragment loaded once
// per K-step and fed to 4 v_wmma_f32_16x16x4_f32. 4 waves / block cover 4
// M-tiles. NC must be a multiple of 64 (256 and 64 here).
__global__ void gemm_wmma_f32(const float* __restrict__ A,
                              const float* __restrict__ B,
                              float* __restrict__ C,
                              int M, int K, int NC) {
    const int lane  = threadIdx.x & 31;
    const int wave  = threadIdx.x >> 5;
    const int tileM = blockIdx.y * 4 + wave;
    if (tileM * 16 >= M) return;
    const int colBase = blockIdx.x * 64;

    const int rowA  = tileM * 16 + (lane & 15);
    const int col   = colBase + (lane & 15);
    const int khalf = (lane >> 4) * 2;   // lanes 0-15: K=0,1 ; lanes 16-31: K=2,3

    v8f acc0 = {0.f,0.f,0.f,0.f,0.f,0.f,0.f,0.f};
    v8f acc1 = acc0, acc2 = acc0, acc3 = acc0;

    for (int k = 0; k < K; k += 4) {
        const int ka = k + khalf;
        v2f a;
        a.x = A[(size_t)rowA * K + ka];
        a.y = A[(size_t)rowA * K + ka + 1];
        const float* b0p = B + (size_t)ka * NC + col;
        const float* b1p = B + (size_t)(ka + 1) * NC + col;
        v2f b0 = { b0p[ 0], b1p[ 0] };
        v2f b1 = { b0p[16], b1p[16] };
        v2f b2 = { b0p[32], b1p[32] };
        v2f b3 = { b0p[48], b1p[48] };
        acc0 = __builtin_amdgcn_wmma_f32_16x16x4_f32(false, a, false, b0, (short)0, acc0, false, false);
        acc1 = __builtin_amdgcn_wmma_f32_16x16x4_f32(false, a, false, b1, (short)0, acc1, false, false);
        acc2 = __builtin_amdgcn_wmma_f32_16x16x4_f32(false, a, false, b2, (short)0, acc2, false, false);
        acc3 = __builtin_amdgcn_wmma_f32_16x16x4_f32(false, a, false, b3, (short)0, acc3, false, false);
    }

    const int crow = tileM * 16 + (lane >> 4) * 8;
#pragma unroll
    for (int r = 0; r < 8; ++r) {
        float* cp = C + (size_t)(crow + r) * NC + col;
        cp[ 0] = acc0[r];
        cp[16] = acc1[r];
        cp[32] = acc2[r];
        cp[48] = acc3[r];
    }
}

// ---------- per-node attention logits: as[n,h] = <xt[n,h,:], a_src[h,:]> ----------
__global__ void alpha_kernel(const float* __restrict__ xt,
                             const float* __restrict__ a_src,
                             const float* __restrict__ a_dst,
                             float* __restrict__ as_, float* __restrict__ ad_,
                             int n, int H, int C) {
    int tid = blockIdx.x * blockDim.x + threadIdx.x;
    if (tid >= n * H) return;
    int node = tid / H, h = tid % H;
    const float4* xp = (const float4*)(xt + (size_t)node * H * C + (size_t)h * C);
    const float4* sp = (const float4*)(a_src + h * C);
    const float4* dp = (const float4*)(a_dst + h * C);
    float ss = 0.f, sd = 0.f;
    for (int c = 0; c < C / 4; ++c) {
        float4 v = xp[c], s4 = sp[c], d4 = dp[c];
        ss += v.x * s4.x + v.y * s4.y + v.z * s4.z + v.w * s4.w;
        sd += v.x * d4.x + v.y * d4.y + v.z * d4.z + v.w * d4.w;
    }
    as_[tid] = ss;
    ad_[tid] = sd;
}

// ---------- pass 1: segment max over dst (flip-trick uint atomicMax) ----------
template <int H, int C>
__global__ void edge_max_kernel(const long long* __restrict__ ei, long long E,
                                long long E2,
                                const float* __restrict__ as_,
                                const float* __restrict__ ad_,
                                unsigned* __restrict__ mu) {
    long long e = (long long)blockIdx.x * blockDim.x + threadIdx.x;
    if (e >= E2) return;
    int s, d; resolve_edge(ei, E, e, s, d);
#pragma unroll
    for (int h = 0; h < H; ++h) {
        float v = as_[s * H + h] + ad_[d * H + h];
        v = v > 0.f ? v : NEG_SLOPE * v;
        atomicMax(&mu[d * H + h], flipf(v));
    }
}

// ---------- pass 2: segment sum of exp(e - max) ----------
template <int H, int C>
__global__ void edge_denom_kernel(const long long* __restrict__ ei, long long E,
                                  long long E2,
                                  const float* __restrict__ as_,
                                  const float* __restrict__ ad_,
                                  const unsigned* __restrict__ mu,
                                  float* __restrict__ denom) {
    long long e = (long long)blockIdx.x * blockDim.x + threadIdx.x;
    if (e >= E2) return;
    int s, d; resolve_edge(ei, E, e, s, d);
#pragma unroll
    for (int h = 0; h < H; ++h) {
        float v = as_[s * H + h] + ad_[d * H + h];
        v = v > 0.f ? v : NEG_SLOPE * v;
        float ex = expf(v - unflipf(mu[d * H + h]));
        unsafeAtomicAdd(&denom[d * H + h], ex);
    }
}

// ---------- pass 3: weighted message scatter (one wave per edge) ----------
// Each lane owns a contiguous V = H*C/32 chunk (never crosses a head since
// V divides C), loaded as float4/float2 -> global_load_b128/b64.
template <int H, int C>
__global__ void edge_msg_kernel(const long long* __restrict__ ei, long long E,
                                long long E2,
                                const float* __restrict__ as_,
                                const float* __restrict__ ad_,
                                const unsigned* __restrict__ mu,
                                const float* __restrict__ denom,
                                const float* __restrict__ xt,
                                float* __restrict__ acc) {
    constexpr int F = H * C;
    constexpr int V = F / 32;            // 8 (layer 1) or 2 (layer 2)
    const int lane = threadIdx.x & 31;
    long long e = ((long long)blockIdx.x * blockDim.x + threadIdx.x) >> 5;
    if (e >= E2) return;
    int s, d; resolve_edge(ei, E, e, s, d);

    float alpha[H];
#pragma unroll
    for (int h = 0; h < H; ++h) {
        float v = as_[s * H + h] + ad_[d * H + h];
        v = v > 0.f ? v : NEG_SLOPE * v;
        float ex = expf(v - unflipf(mu[d * H + h]));
        alpha[h] = ex / (denom[d * H + h] + 1e-16f);
    }

    const int j0 = lane * V;
    const float a = alpha[j0 / C];       // chunk lies within one head
    const float* xs = xt + (size_t)s * F + j0;
    float* ap = acc + (size_t)d * F + j0;

    float vals[V];
    if constexpr (V == 8) {
        float4 p0 = ((const float4*)xs)[0];
        float4 p1 = ((const float4*)xs)[1];
        vals[0]=p0.x; vals[1]=p0.y; vals[2]=p0.z; vals[3]=p0.w;
        vals[4]=p1.x; vals[5]=p1.y; vals[6]=p1.z; vals[7]=p1.w;
    } else {
        float2 p0 = ((const float2*)xs)[0];
        vals[0]=p0.x; vals[1]=p0.y;
    }
#pragma unroll
    for (int v = 0; v < V; ++v)
        unsafeAtomicAdd(&ap[v], a * vals[v]);
}

// ---------- ELU(in-place) + bias ----------
__global__ void elu_bias_kernel(float* __restrict__ buf,
                                const float* __restrict__ bias, long long total, int F) {
    long long tid = (long long)blockIdx.x * blockDim.x + threadIdx.x;
    if (tid >= total) return;
    float v = buf[tid] + bias[tid % F];
    buf[tid] = v > 0.f ? v : (expf(v) - 1.f);
}

// ---------- final bias add ----------
__global__ void bias_kernel(float* __restrict__ buf,
                            const float* __restrict__ bias, long long total, int F) {
    long long tid = (long long)blockIdx.x * blockDim.x + threadIdx.x;
    if (tid >= total) return;
    buf[tid] += bias[tid % F];
}

extern "C" void kernel_launch(void* const* d_in, const int* in_sizes, int n_in,
                              void* d_out, int out_size, void* d_ws, size_t ws_size,
                              hipStream_t stream) {
    const float*     x    = (const float*)d_in[0];
    const long long* ei   = (const long long*)d_in[1];
    const float*     W1   = (const float*)d_in[2];
    const float*     as1w = (const float*)d_in[3];
    const float*     ad1w = (const float*)d_in[4];
    const float*     b1   = (const float*)d_in[5];
    const float*     W2   = (const float*)d_in[6];
    const float*     as2w = (const float*)d_in[7];
    const float*     ad2w = (const float*)d_in[8];
    const float*     b2   = (const float*)d_in[9];
    float* out = (float*)d_out;

    const int  N  = in_sizes[0] / IN_C;        // 50000
    const long long E  = in_sizes[1] / 2;      // 800000
    const long long E2 = E + N;                // edges + self loops
    const int F1 = HEADS * HID;                // 256

    // workspace carve-up (floats)
    float* ws = (float*)d_ws;
    size_t off = 0;
    float*    xt1  = ws + off; off += (size_t)N * F1;
    float*    acc1 = ws + off; off += (size_t)N * F1;   // becomes h after ELU
    float*    xt2  = ws + off; off += (size_t)N * OUT_C;
    float*    as1  = ws + off; off += (size_t)N * HEADS;
    float*    ad1  = ws + off; off += (size_t)N * HEADS;
    float*    den1 = ws + off; off += (size_t)N * HEADS;
    unsigned* m1   = (unsigned*)(ws + off); off += (size_t)N * HEADS;
    float*    as2  = ws + off; off += (size_t)N;
    float*    ad2  = ws + off; off += (size_t)N;
    float*    den2 = ws + off; off += (size_t)N;
    unsigned* m2   = (unsigned*)(ws + off); off += (size_t)N;

    // zero-init (graph-capture safe). flip-space min is 0 -> memset(0) works for max buf.
    hipMemsetAsync(acc1, 0, (size_t)N * F1 * sizeof(float), stream);
    hipMemsetAsync(den1, 0, (size_t)N * HEADS * sizeof(float), stream);
    hipMemsetAsync(m1,   0, (size_t)N * HEADS * sizeof(unsigned), stream);
    hipMemsetAsync(den2, 0, (size_t)N * sizeof(float), stream);
    hipMemsetAsync(m2,   0, (size_t)N * sizeof(unsigned), stream);
    hipMemsetAsync(out,  0, (size_t)out_size * sizeof(float), stream);

    const int TB = 256;
    const long long eb  = (E2 + TB - 1) / TB;       // thread-per-edge blocks
    const long long ebw = (E2 * 32 + TB - 1) / TB;  // wave-per-edge blocks

    // ===== layer 1 =====
    {
        dim3 grid(F1 / 64, (N / 16 + 3) / 4);
        gemm_wmma_f32<<<grid, 128, 0, stream>>>(x, W1, xt1, N, IN_C, F1);
    }
    alpha_kernel<<<(N * HEADS + TB - 1) / TB, TB, 0, stream>>>(xt1, as1w, ad1w, as1, ad1, N, HEADS, HID);
    edge_max_kernel<HEADS, HID><<<(int)eb, TB, 0, stream>>>(ei, E, E2, as1, ad1, m1);
    edge_denom_kernel<HEADS, HID><<<(int)eb, TB, 0, stream>>>(ei, E, E2, as1, ad1, m1, den1);
    edge_msg_kernel<HEADS, HID><<<(int)ebw, TB, 0, stream>>>(ei, E, E2, as1, ad1, m1, den1, xt1, acc1);
    elu_bias_kernel<<<(int)(((long long)N * F1 + TB - 1) / TB), TB, 0, stream>>>(acc1, b1, (long long)N * F1, F1);

    // ===== layer 2 (1 head, concat=False -> mean == identity) =====
    {
        dim3 grid(OUT_C / 64, (N / 16 + 3) / 4);
        gemm_wmma_f32<<<grid, 128, 0, stream>>>(acc1, W2, xt2, N, F1, OUT_C);
    }
    alpha_kernel<<<(N + TB - 1) / TB, TB, 0, stream>>>(xt2, as2w, ad2w, as2, ad2, N, 1, OUT_C);
    edge_max_kernel<1, OUT_C><<<(int)eb, TB, 0, stream>>>(ei, E, E2, as2, ad2, m2);
    edge_denom_kernel<1, OUT_C><<<(int)eb, TB, 0, stream>>>(ei, E, E2, as2, ad2, m2, den2);
    edge_msg_kernel<1, OUT_C><<<(int)ebw, TB, 0, stream>>>(ei, E, E2, as2, ad2, m2, den2, xt2, out);
    bias_kernel<<<(int)(((long long)N * OUT_C + TB - 1) / TB), TB, 0, stream>>>(out, b2, (long long)N * OUT_C, OUT_C);
}